// GCN_20383914786987
// MI455X (gfx1250) — compile-verified
//
#include <hip/hip_runtime.h>
#include <hip/hip_bf16.h>

// GCN 2-layer forward for MI455X (gfx1250, wave32, WMMA).
//   pre1 = X @ W1                      (WMMA f32 16x16x4)
//   agg1 = A @ pre1                    (edge-parallel gather + f32 atomic scatter, L2-resident)
//   pre2 = relu(agg1) @ W2             (WMMA f32 16x16x4, relu fused into A-load)
//   out  = A @ pre2                    (edge-parallel gather + f32 atomic scatter)

#define N_NODES 100000
#define N_EDGES 3200000
#define IN_DIM  256
#define HIDDEN  128
#define OUT_DIM 64

typedef __attribute__((ext_vector_type(2))) float v2f;
typedef __attribute__((ext_vector_type(8))) float v8f;

// ---------------------------------------------------------------------------
// GEMM1: pre1[N,128] = X[N,256] @ W1[256,128]
// Block = 256 threads = 8 waves; block handles one 16-row strip; wave w owns
// the 16-col tile starting at w*16 (8*16 == 128 covers all of HIDDEN).
// ---------------------------------------------------------------------------
__global__ __launch_bounds__(256) void gcn_gemm1(const float* __restrict__ X,
                                                 const float* __restrict__ W1,
                                                 float* __restrict__ pre1) {
    const int lane  = threadIdx.x & 31;
    const int wave  = threadIdx.x >> 5;
    const int mBase = blockIdx.x * 16;
    const int nBase = wave * 16;

    const int mRow = mBase + (lane & 15);     // A fragment: M = lane%16
    const int kOff = (lane >> 4) * 2;         // A/B fragment: K = 2*(lane/16) + v
    const int nCol = nBase + (lane & 15);     // B/C fragment: N = lane%16

    const float* __restrict__ xr = X + (size_t)mRow * IN_DIM;

    v8f c = {};
    for (int k = 0; k < IN_DIM; k += 4) {
        v2f a, b;
        a.x = xr[k + kOff];
        a.y = xr[k + kOff + 1];
        b.x = W1[(size_t)(k + kOff)     * HIDDEN + nCol];
        b.y = W1[(size_t)(k + kOff + 1) * HIDDEN + nCol];
        c = __builtin_amdgcn_wmma_f32_16x16x4_f32(false, a, false, b,
                                                  (short)0, c, false, false);
    }

    // C/D layout: VGPR v, lanes 0-15 -> M=v, lanes 16-31 -> M=v+8; N=lane%16
    const int mHalf = (lane >> 4) * 8;
#pragma unroll
    for (int v = 0; v < 8; ++v) {
        pre1[(size_t)(mBase + mHalf + v) * HIDDEN + nCol] = c[v];
    }
}

// ---------------------------------------------------------------------------
// GEMM2: pre2[N,64] = relu(agg1)[N,128] @ W2[128,64]
// Block = 8 waves: 2 M-tiles x 4 N-tiles (4*16 == 64 covers OUT_DIM).
// ---------------------------------------------------------------------------
__global__ __launch_bounds__(256) void gcn_gemm2(const float* __restrict__ AGG,
                                                 const float* __restrict__ W2,
                                                 float* __restrict__ pre2) {
    const int lane  = threadIdx.x & 31;
    const int wave  = threadIdx.x >> 5;
    const int mBase = (blockIdx.x * 2 + (wave >> 2)) * 16;
    const int nBase = (wave & 3) * 16;

    const int mRow = mBase + (lane & 15);
    const int kOff = (lane >> 4) * 2;
    const int nCol = nBase + (lane & 15);

    const float* __restrict__ ar = AGG + (size_t)mRow * HIDDEN;

    v8f c = {};
    for (int k = 0; k < HIDDEN; k += 4) {
        v2f a, b;
        a.x = fmaxf(ar[k + kOff], 0.0f);          // fused ReLU
        a.y = fmaxf(ar[k + kOff + 1], 0.0f);
        b.x = W2[(size_t)(k + kOff)     * OUT_DIM + nCol];
        b.y = W2[(size_t)(k + kOff + 1) * OUT_DIM + nCol];
        c = __builtin_amdgcn_wmma_f32_16x16x4_f32(false, a, false, b,
                                                  (short)0, c, false, false);
    }

    const int mHalf = (lane >> 4) * 8;
#pragma unroll
    for (int v = 0; v < 8; ++v) {
        pre2[(size_t)(mBase + mHalf + v) * OUT_DIM + nCol] = c[v];
    }
}

// ---------------------------------------------------------------------------
// SpMM scatter: out[dst[e]] += vals[e] * H[src[e]], D = 128 or 64.
// One wave per edge; lane loads D/32 consecutive floats (float4 / float2,
// 128b/64b coalesced), scales, and scatters with global_atomic_add_f32.
// Feature matrices (<=51.2 MB) and destinations are L2-resident (192 MB L2).
// ---------------------------------------------------------------------------
template <int D>
__global__ __launch_bounds__(256) void gcn_spmm(const float* __restrict__ H,
                                                const float* __restrict__ vals,
                                                const int* __restrict__ src,
                                                const int* __restrict__ dst,
                                                float* __restrict__ out) {
    constexpr int V = D / 32;                   // floats per lane (4 or 2)
    const int lane   = threadIdx.x & 31;
    const int wid    = (blockIdx.x * blockDim.x + threadIdx.x) >> 5;
    const int nWaves = (gridDim.x * blockDim.x) >> 5;

    for (int e = wid; e < N_EDGES; e += nWaves) {
        const int   s = src[e];
        const int   d = dst[e];
        const float w = vals[e];

        const float* __restrict__ hp = H + (size_t)s * D + lane * V;
        float* __restrict__ op = out + (size_t)d * D + lane * V;

        if constexpr (V == 4) {
            const float4 hv = *reinterpret_cast<const float4*>(hp);
            atomicAdd(op + 0, w * hv.x);
            atomicAdd(op + 1, w * hv.y);
            atomicAdd(op + 2, w * hv.z);
            atomicAdd(op + 3, w * hv.w);
        } else {
            const float2 hv = *reinterpret_cast<const float2*>(hp);
            atomicAdd(op + 0, w * hv.x);
            atomicAdd(op + 1, w * hv.y);
        }
    }
}

// ---------------------------------------------------------------------------
extern "C" void kernel_launch(void* const* d_in, const int* in_sizes, int n_in,
                              void* d_out, int out_size, void* d_ws, size_t ws_size,
                              hipStream_t stream) {
    const float* x    = (const float*)d_in[0];   // [N_NODES, IN_DIM]
    const float* vals = (const float*)d_in[1];   // [N_EDGES]
    const float* w1   = (const float*)d_in[2];   // [IN_DIM, HIDDEN]
    const float* w2   = (const float*)d_in[3];   // [HIDDEN, OUT_DIM]
    const int*   esrc = (const int*)d_in[4];     // [N_EDGES]
    const int*   edst = (const int*)d_in[5];     // [N_EDGES]
    float*       out  = (float*)d_out;           // [N_NODES, OUT_DIM]

    float* pre1 = (float*)d_ws;                           // N*128 f32 = 51.2 MB
    float* agg1 = pre1 + (size_t)N_NODES * HIDDEN;        // N*128 f32 = 51.2 MB
    float* pre2 = agg1 + (size_t)N_NODES * HIDDEN;        // N*64  f32 = 25.6 MB

    // Zero the atomic accumulation targets (graph-capture safe).
    hipMemsetAsync(agg1, 0, (size_t)N_NODES * HIDDEN * sizeof(float), stream);
    hipMemsetAsync(out,  0, (size_t)N_NODES * OUT_DIM * sizeof(float), stream);

    // Layer 1 dense: pre1 = X @ W1   (100000/16 = 6250 row strips)
    gcn_gemm1<<<N_NODES / 16, 256, 0, stream>>>(x, w1, pre1);

    // Layer 1 sparse: agg1 = A @ pre1
    gcn_spmm<HIDDEN><<<4096, 256, 0, stream>>>(pre1, vals, esrc, edst, agg1);

    // Layer 2 dense: pre2 = relu(agg1) @ W2   (2 M-tiles per block)
    gcn_gemm2<<<N_NODES / 32, 256, 0, stream>>>(agg1, w2, pre2);

    // Layer 2 sparse: out = A @ pre2
    gcn_spmm<OUT_DIM><<<4096, 256, 0, stream>>>(pre2, vals, esrc, edst, out);
}